// Subtraction_59966333386861
// MI455X (gfx1250) — compile-verified
//
#include <hip/hip_runtime.h>
#include <stdint.h>

typedef float v4f  __attribute__((ext_vector_type(4)));
typedef unsigned int u32x4 __attribute__((ext_vector_type(4)));
typedef int  i32x4 __attribute__((ext_vector_type(4)));
typedef int  i32x8 __attribute__((ext_vector_type(8)));

#define HW       112
#define PITCH    120          // floats per LDS row (bank-skewed, 16B aligned)
#define ROWS     14           // 8 output rows + 3 halo above + 3 below
#define BAND     8
#define NTHREADS 224          // 7 waves of 32

__device__ __forceinline__ int refl(int i) {
  // jnp.pad 'reflect' (mirror, edge not repeated), H=W=112
  return i < 0 ? -i : (i > 111 ? 222 - i : i);
}

__global__ __launch_bounds__(NTHREADS)
void subtraction_k7_kernel(const float* __restrict__ x, float* __restrict__ out) {
  __shared__ float smem[ROWS * PITCH];   // 6720 B of 320 KB/WGP

  const int tid  = threadIdx.x;
  const int band = blockIdx.x;           // 0..13
  const int chan = blockIdx.y;           // 0..255 == n*64 + c
  const int h0   = band * BAND;

  // contiguous valid global rows covering [h0-3, h0+10]
  const int g_lo    = max(0, h0 - 3);
  const int g_hi    = min(111, h0 + 10);
  const int nrows   = g_hi - g_lo + 1;
  const int r_start = g_lo - (h0 - 3);   // LDS row holding global row g_lo

  const float* xc = x + (size_t)chan * (HW * HW);

#if defined(__has_builtin) && __has_builtin(__builtin_amdgcn_tensor_load_to_lds)
  // --- Tensor Data Mover: DMA nrows x 112 fp32 into pitch-120 LDS rows,
  //     interior columns land at column offset +3 (halo columns added later).
  if (tid < 32) {                        // wave 0 issues; EXEC ignored by TDM
    uint64_t gaddr = (uint64_t)(uintptr_t)(xc + (size_t)g_lo * HW);
    uint32_t laddr = (uint32_t)(uintptr_t)(&smem[r_start * PITCH + 3]);

    u32x4 g0; i32x8 g1; i32x4 g2; i32x4 g3; i32x8 gz;
    // D# group 0: count=1 | lds_addr | global_addr[56:0] | type=2
    g0[0] = 1u;
    g0[1] = laddr;
    g0[2] = (uint32_t)gaddr;
    g0[3] = (uint32_t)((gaddr >> 32) & 0x01FFFFFFu) | (2u << 30);
    // D# group 1: mask=0, data_size=4B(2), iterate_enable=1
    g1[0] = (2 << 16) | (1 << 19);
    g1[1] = (HW & 0xFFFF) << 16;        // tensor_dim0 = 112 (low16)
    g1[2] = (nrows & 0xFFFF) << 16;     // tensor_dim0 hi=0 | tensor_dim1 lo16
    g1[3] = (HW << 16);                 // tensor_dim1 hi=0 | tile_dim0 = 112
    g1[4] = 1;                          // tile_dim1 = 1 row per iteration
    g1[5] = HW;                         // tensor_dim0_stride = 112 (lo32)
    g1[6] = (HW & 0xFFFF) << 16;        // dim0_stride hi=0 | dim1_stride lo16
    g1[7] = 0;
    // D# group 2 (iterate mode): lds_incr=120 elems, global_incr=112 elems
    g2[0] = 1;                          // tensor_dim2 (unused)
    g2[1] = PITCH;                      // lds_addr_increment
    g2[2] = HW;                         // global_addr_increment (lo32)
    g2[3] = (nrows - 1) << 16;          // incr hi=0 | iterate_count = nrows-1
    g3[0] = 0; g3[1] = 0; g3[2] = 0; g3[3] = 0;
    gz[0] = 0; gz[1] = 0; gz[2] = 0; gz[3] = 0;
    gz[4] = 0; gz[5] = 0; gz[6] = 0; gz[7] = 0;

    // 6-arg form on this toolchain: groups 0..3, reserved vector, cpol
    __builtin_amdgcn_tensor_load_to_lds(g0, g1, g2, g3, gz, 0);
    __builtin_amdgcn_s_wait_tensorcnt(0);
  }
#else
  for (int idx = tid; idx < nrows * HW; idx += NTHREADS) {
    int r = idx / HW, c = idx % HW;
    smem[(r_start + r) * PITCH + 3 + c] = xc[(size_t)(g_lo + r) * HW + c];
  }
#endif
  __syncthreads();

  // --- row-reflection fixups (only first/last band): LDS -> LDS copies
  if (nrows < ROWS) {
    for (int idx = tid; idx < ROWS * HW; idx += NTHREADS) {
      int r  = idx / HW;
      int gr = h0 - 3 + r;
      if (gr >= 0 && gr <= 111) continue;
      int sr = refl(gr) - (h0 - 3);           // source LDS row (always valid)
      int c  = idx % HW;
      smem[r * PITCH + 3 + c] = smem[sr * PITCH + 3 + c];
    }
    __syncthreads();
  }

  // --- reflected column halos: cols 0..2 and 115..117 (84 copies, 1 pass)
  if (tid < ROWS * 6) {
    int r   = tid / 6, j = tid % 6;
    int dst = (j < 3) ? j : (112 + j);        // 0,1,2,115,116,117
    int src = (j < 3) ? (6 - j) : (228 - (112 + j));
    smem[r * PITCH + dst] = smem[r * PITCH + src];
  }
  __syncthreads();

  // --- compute: one float4 of one output row per thread, 49 NT stores
  const int hr = tid / 28;                    // 0..7
  const int w  = (tid % 28) * 4;              // 0..108, float4-aligned
  const int h  = h0 + hr;

  const float* crow = &smem[(hr + 3) * PITCH + (w + 3)];
  v4f center;
  center[0] = crow[0]; center[1] = crow[1];
  center[2] = crow[2]; center[3] = crow[3];

  float* obase = out + (size_t)chan * (49u * HW * HW) + (size_t)h * HW + w;

#pragma unroll
  for (int kh = 0; kh < 7; ++kh) {
    // 12-float window (3 aligned ds_load_b128), covers all 7 kw shifts
    const v4f* lrow = (const v4f*)&smem[(hr + kh) * PITCH + w];
    v4f a = lrow[0], b = lrow[1], c = lrow[2];
    float win[12] = {a[0], a[1], a[2], a[3],
                     b[0], b[1], b[2], b[3],
                     c[0], c[1], c[2], c[3]};
#pragma unroll
    for (int kw = 0; kw < 7; ++kw) {
      v4f nb = {win[kw], win[kw + 1], win[kw + 2], win[kw + 3]};
      v4f o  = center - nb;
      // streaming output (write-once, > L2): non-temporal b128 store
      __builtin_nontemporal_store(
          o, (v4f*)(obase + (size_t)(kh * 7 + kw) * (HW * HW)));
    }
  }
}

extern "C" void kernel_launch(void* const* d_in, const int* in_sizes, int n_in,
                              void* d_out, int out_size, void* d_ws, size_t ws_size,
                              hipStream_t stream) {
  (void)in_sizes; (void)n_in; (void)out_size; (void)d_ws; (void)ws_size;
  const float* x   = (const float*)d_in[0];   // (4,64,112,112) fp32
  float*       out = (float*)d_out;           // (4,64,49,12544) fp32
  dim3 grid(14, 256, 1);                      // 14 row-bands x (N*C)
  subtraction_k7_kernel<<<grid, NTHREADS, 0, stream>>>(x, out);
}